// Movie_Recommendation_Model_15436112461945
// MI455X (gfx1250) — compile-verified
//
#include <hip/hip_runtime.h>
#include <hip/hip_bf16.h>

typedef __attribute__((ext_vector_type(2))) float v2f;
typedef __attribute__((ext_vector_type(8))) float v8f;

namespace {

// ---------------- LDS arena (units: floats), ~60 KB peak -----------------
// Liveness:
//  phase 1 (title):  P(write) + TC(2x960 double buffer) + CW
//  phase 2 (movie heads): P(read) + M + E
//  phase 3 (movie_feature): M + UM (+ async user gathers into E)
//  phase 4 (user):   UM + U + E
//  phase 5 (rate):   UM + H
constexpr int OFF_P  = 0;       // 16*720 = 11520  pooled conv features (phases 1-2)
constexpr int OFF_UM = 0;       // 16*400 = 6400   user||movie concat (alias P, phases 3-5)
constexpr int OFF_H  = 6400;    // 16*128 = 2048   rate hidden (alias P tail, phase 5)
constexpr int OFF_U  = 8448;    // 16*128 = 2048   user concat (alias P tail, phase 4)
constexpr int OFF_M  = 11520;   // 16*96  = 1536   movie concat (phases 2-3)
constexpr int OFF_TC = 13056;   // 2*960  = 1920   title-emb double buffer (phase 1 only)
constexpr int OFF_E  = 13056;   // up to 1280      embedding staging (alias TC, phases 2-4)
constexpr int OFF_CW = 14976;   // 72   conv weights
constexpr int OFF_CB = 15048;   // 8    conv bias
constexpr int LDS_FLOATS = 15056;   // 60224 bytes -> ~5 workgroups (10 waves) per WGP

struct MRParams {
  const int   *uid, *gender, *age, *job, *mid, *cats, *titles;
  const float *uid_tab, *gender_tab, *age_tab, *job_tab, *mid_tab, *cat_tab, *title_tab;
  const float *uid_W, *uid_b, *g_W, *g_b, *a_W, *a_b, *j_W, *j_b;
  const float *uf_W, *uf_b;
  const float *mid_W, *mid_b, *cat_W, *cat_b, *tf_W, *tf_b, *mf_W, *mf_b;
  const float *r1_W, *r1_b, *r2_W, *r2_b;
  const float *conv_w, *conv_b;
  float *out;
  int batch;
};

// ---- CDNA5 async global->LDS copy (16B per lane), tracked by ASYNCcnt ----
// VDST = LDS byte address (low 32 bits of generic LDS pointer, per flat aperture
// mapping); ADDR = 64-bit global address.
__device__ __forceinline__ void async_b128(void* lds_dst, const void* gsrc) {
  unsigned l = (unsigned)(unsigned long long)lds_dst;
  asm volatile("global_load_async_to_lds_b128 %0, %1, off"
               :: "v"(l), "v"(gsrc) : "memory");
}
__device__ __forceinline__ void wait_async0() {
  asm volatile("s_wait_asynccnt 0x0" ::: "memory");
}

// ---------------- WMMA f32 16x16x4 GEMM helper -----------------
// OUT[16][Nout] = A[16][K] * W^T + bias  (W row-major [Nout][K], global/L2-hot)
// N-tiles n0 = (tile0 + k*tstride)*16 are handled by this wave (wave-uniform split).
// A fragment (32-bit A 16x4):   lane<16 -> row=lane, K=k0,k0+1 ; lane>=16 -> K=k0+2,k0+3
// B fragment (4x16 = W^T tile): lane<16 -> col=lane, K=k0,k0+1 ; lane>=16 -> K=k0+2,k0+3
// C/D (16x16 f32): vgpr r -> row r + 8*(lane>=16), col = lane&15
__device__ __forceinline__ void fc16(const float* __restrict__ A, int lda, int K,
                                     const float* __restrict__ W,
                                     const float* __restrict__ bias, int Nout,
                                     float* __restrict__ OUT, int ldo,
                                     int lane, bool relu, int tile0, int tstride) {
  const int l    = lane & 15;
  const int half = lane >> 4;
  const float* Ar = A + (long)l * lda + half * 2;
  for (int n0 = tile0 * 16; n0 < Nout; n0 += tstride * 16) {
    const int  n     = n0 + l;
    const bool valid = (n < Nout);
    const int  nc    = valid ? n : (Nout - 1);   // clamp so loads stay in-bounds
    const float msk  = valid ? 1.0f : 0.0f;      // zero-mask: no EXEC divergence at WMMA
    const float* Wp  = W + (long)nc * K + half * 2;
    v8f acc = {};
    for (int k0 = 0; k0 < K; k0 += 4) {
      v2f a, b;
      a.x = Ar[k0];
      a.y = Ar[k0 + 1];
      b.x = Wp[k0] * msk;
      b.y = Wp[k0 + 1] * msk;
      acc = __builtin_amdgcn_wmma_f32_16x16x4_f32(
          /*neg_a=*/false, a, /*neg_b=*/false, b,
          /*c_mod=*/(short)0, acc, /*reuse_a=*/false, /*reuse_b=*/false);
    }
    if (valid) {
      const float bv = bias[nc];
      #pragma unroll
      for (int r8 = 0; r8 < 8; ++r8) {
        float v = acc[r8] + bv;
        if (relu) v = fmaxf(v, 0.0f);
        OUT[(r8 + half * 8) * ldo + n] = v;
      }
    }
  }
}

__global__ void __launch_bounds__(64)
movie_rec_kernel(MRParams p) {
  extern __shared__ float lds[];
  const int tid  = threadIdx.x;        // 64 threads = 2 waves per 16-row tile
  const int lane = tid & 31;
  const int wv   = tid >> 5;           // wave id 0/1 (wave-uniform)
  const int row0 = blockIdx.x * 16;

  float* P  = lds + OFF_P;
  float* UM = lds + OFF_UM;
  float* H  = lds + OFF_H;
  float* U  = lds + OFF_U;
  float* M  = lds + OFF_M;
  float* TC = lds + OFF_TC;
  float* E  = lds + OFF_E;
  float* CW = lds + OFF_CW;
  float* CB = lds + OFF_CB;

  // ---- phase 0: stage conv weights ----
  if (tid < 72) CW[tid] = p.conv_w[tid];
  if (tid < 8)  CB[tid] = p.conv_b[tid];

  // ---- phase 1: title branch, 2 rows/chunk, double-buffered async gathers ----
  // conv 3x3 (1->8ch, VALID on 15x32) + 2x2/2 maxpool -> P[16][720]
  // pooled flat order: ch*90 + py*15 + px  (matches reshape of [8,6,15])
  float* TB0 = TC;
  float* TB1 = TC + 960;
  auto issue_title_chunk = [&](int c, float* buf) {
    const int rbase = c * 2;
    for (int j = tid; j < 2 * 15 * 8; j += 64) {   // 240 x b128 transfers
      const int slot = j >> 3, q = j & 7;
      const int r = slot / 15, w = slot - r * 15;
      const int id = p.titles[(row0 + rbase + r) * 15 + w];
      async_b128(buf + r * 480 + w * 32 + q * 4,
                 p.title_tab + (long)id * 32 + q * 4);
    }
  };
  issue_title_chunk(0, TB0);
  for (int c = 0; c < 8; ++c) {
    float* buf = (c & 1) ? TB1 : TB0;
    wait_async0();        // chunk c resident (this wave's copies)
    __syncthreads();      // all waves' copies done; prev conv readers done
    if (c + 1 < 8) issue_title_chunk(c + 1, (c & 1) ? TB0 : TB1);  // overlap with conv
    const int rbase = c * 2;
    for (int i = tid; i < 2 * 720; i += 64) {      // fused conv + maxpool
      const int r  = i / 720;
      const int t  = i - r * 720;
      const int ch = t / 90;
      const int q  = t - ch * 90;
      const int py = q / 15;
      const int px = q - py * 15;
      const float* img = buf + r * 480;
      const float* w9  = CW + ch * 9;
      const float  cb  = CB[ch];
      float m = -3.402823466e38f;
      #pragma unroll
      for (int yy = 0; yy < 2; ++yy) {
        #pragma unroll
        for (int xx = 0; xx < 2; ++xx) {
          const int y = 2 * py + yy, x = 2 * px + xx;
          float s = cb;
          #pragma unroll
          for (int dy = 0; dy < 3; ++dy)
            #pragma unroll
            for (int dx = 0; dx < 3; ++dx)
              s = fmaf(img[(y + dy) * 32 + (x + dx)], w9[dy * 3 + dx], s);
          m = fmaxf(m, s);
        }
      }
      P[(rbase + r) * 720 + t] = m;
    }
  }
  __syncthreads();   // title pipeline complete; TC region now free (-> E)

  // ---- phase 2: movie branch heads -> M[16][96] ----
  // stage: mid emb (async) -> E[0:512), cat mean (VALU) -> E[512:1024)
  float* Emid = E;
  float* Ecat = E + 512;
  for (int i = tid; i < 16 * 8; i += 64) {
    const int r = i >> 3, q = i & 7;
    async_b128(Emid + r * 32 + q * 4,
               p.mid_tab + (long)p.mid[row0 + r] * 32 + q * 4);
  }
  for (int i = tid; i < 16 * 32; i += 64) {
    const int r = i >> 5, cc = i & 31;
    const int* cr = p.cats + (row0 + r) * 18;
    float s = 0.0f;
    #pragma unroll
    for (int w = 0; w < 18; ++w) s += p.cat_tab[cr[w] * 32 + cc];
    Ecat[r * 32 + cc] = s * (1.0f / 18.0f);
  }
  wait_async0();
  __syncthreads();
  // wave 0: mid_fc (32->32); wave 1: cat_fc (32->32) — concurrent
  if (wv == 0) fc16(Emid, 32, 32, p.mid_W, p.mid_b, 32, M + 0,  96, lane, false, 0, 1);
  else         fc16(Ecat, 32, 32, p.cat_W, p.cat_b, 32, M + 32, 96, lane, false, 0, 1);
  // title_fc: [16x720]x[720x32], one N-tile per wave
  fc16(P, 720, 720, p.tf_W, p.tf_b, 32, M + 64, 96, lane, false, wv, 2);
  __syncthreads();

  // ---- phase 3: movie_feature [16x96]x[96x200] -> UM[:,200:400] + global ----
  fc16(M, 96, 96, p.mf_W, p.mf_b, 200, UM + 200, 400, lane, false, wv, 2);
  __syncthreads();

  // issue user-branch async gathers now: they overlap the movie copy-out below.
  // uid E[0:512), gender E[512:768), age E[768:1024), job E[1024:1280)
  float* Euid = E;
  float* Egen = E + 512;
  float* Eage = E + 768;
  float* Ejob = E + 1024;
  for (int i = tid; i < 16 * 8; i += 64) {
    const int r = i >> 3, q = i & 7;
    async_b128(Euid + r * 32 + q * 4,
               p.uid_tab + (long)p.uid[row0 + r] * 32 + q * 4);
  }
  for (int i = tid; i < 16 * 4; i += 64) {
    const int r = i >> 2, q = i & 3;
    async_b128(Egen + r * 16 + q * 4, p.gender_tab + (long)p.gender[row0 + r] * 16 + q * 4);
    async_b128(Eage + r * 16 + q * 4, p.age_tab    + (long)p.age[row0 + r]    * 16 + q * 4);
    async_b128(Ejob + r * 16 + q * 4, p.job_tab    + (long)p.job[row0 + r]    * 16 + q * 4);
  }
  // movie output copy-out (overlaps the async gathers above)
  for (int r = 0; r < 16; ++r)
    for (int c = tid; c < 200; c += 64)
      p.out[(long)p.batch * 200 + (long)(row0 + r) * 200 + c] = UM[r * 400 + 200 + c];
  wait_async0();
  __syncthreads();

  // ---- phase 4: user branch -> U[16][128] -> user_feature -> UM[:,0:200] ----
  // wave 0: uid_fc + age_fc; wave 1: gender_fc + job_fc — concurrent
  if (wv == 0) {
    fc16(Euid, 32, 32, p.uid_W, p.uid_b, 32, U + 0,  128, lane, false, 0, 1);
    fc16(Eage, 16, 16, p.a_W,   p.a_b,   32, U + 64, 128, lane, false, 0, 1);
  } else {
    fc16(Egen, 16, 16, p.g_W,   p.g_b,   32, U + 32, 128, lane, false, 0, 1);
    fc16(Ejob, 16, 16, p.j_W,   p.j_b,   32, U + 96, 128, lane, false, 0, 1);
  }
  __syncthreads();
  // user_feature: [16x128]x[128x200], N-tiles split across waves
  fc16(U, 128, 128, p.uf_W, p.uf_b, 200, UM + 0, 400, lane, false, wv, 2);
  __syncthreads();
  for (int r = 0; r < 16; ++r)
    for (int c = tid; c < 200; c += 64)
      p.out[(long)(row0 + r) * 200 + c] = UM[r * 400 + c];
  __syncthreads();

  // ---- phase 5: rating head ----
  // rate_fc1: [16x400]x[400x128] + ReLU -> H, N-tiles split across waves
  fc16(UM, 400, 400, p.r1_W, p.r1_b, 128, H, 128, lane, true, wv, 2);
  __syncthreads();

  // rate_fc2: [16x128].[128] -> scalar per row
  if (tid < 16) {
    const float* h = H + tid * 128;
    float s = p.r2_b[0];
    #pragma unroll 4
    for (int c = 0; c < 128; ++c) s = fmaf(h[c], p.r2_W[c], s);
    p.out[(long)p.batch * 400 + (row0 + tid)] = s;
  }
}

} // anonymous namespace

extern "C" void kernel_launch(void* const* d_in, const int* in_sizes, int n_in,
                              void* d_out, int out_size, void* d_ws, size_t ws_size,
                              hipStream_t stream) {
  (void)n_in; (void)out_size; (void)d_ws; (void)ws_size;
  MRParams p;
  // inputs (setup_inputs dict order)
  p.uid        = (const int*)d_in[0];
  p.gender     = (const int*)d_in[1];
  p.age        = (const int*)d_in[2];
  p.job        = (const int*)d_in[3];
  p.mid        = (const int*)d_in[4];
  p.cats       = (const int*)d_in[5];
  p.titles     = (const int*)d_in[6];
  // params dict order, tuples expanded (W, b)
  p.uid_tab    = (const float*)d_in[7];
  p.gender_tab = (const float*)d_in[8];
  p.age_tab    = (const float*)d_in[9];
  p.job_tab    = (const float*)d_in[10];
  p.mid_tab    = (const float*)d_in[11];
  p.cat_tab    = (const float*)d_in[12];
  p.title_tab  = (const float*)d_in[13];
  p.uid_W = (const float*)d_in[14];  p.uid_b = (const float*)d_in[15];
  p.g_W   = (const float*)d_in[16];  p.g_b   = (const float*)d_in[17];
  p.a_W   = (const float*)d_in[18];  p.a_b   = (const float*)d_in[19];
  p.j_W   = (const float*)d_in[20];  p.j_b   = (const float*)d_in[21];
  p.uf_W  = (const float*)d_in[22];  p.uf_b  = (const float*)d_in[23];
  p.mid_W = (const float*)d_in[24];  p.mid_b = (const float*)d_in[25];
  p.cat_W = (const float*)d_in[26];  p.cat_b = (const float*)d_in[27];
  p.tf_W  = (const float*)d_in[28];  p.tf_b  = (const float*)d_in[29];
  p.mf_W  = (const float*)d_in[30];  p.mf_b  = (const float*)d_in[31];
  p.r1_W  = (const float*)d_in[32];  p.r1_b  = (const float*)d_in[33];
  p.r2_W  = (const float*)d_in[34];  p.r2_b  = (const float*)d_in[35];
  p.conv_w = (const float*)d_in[36];
  p.conv_b = (const float*)d_in[37];
  p.out   = (float*)d_out;
  p.batch = in_sizes[0];

  const int tiles = p.batch / 16;
  const size_t shmem = (size_t)LDS_FLOATS * sizeof(float);
  movie_rec_kernel<<<dim3(tiles), dim3(64), shmem, stream>>>(p);
}